// DEChannelPool_34617436405973
// MI455X (gfx1250) — compile-verified
//
#include <hip/hip_runtime.h>

// ---------------- problem constants ----------------
static constexpr int Bn = 8, Cc = 64, Hh = 192, Ww = 192;
static constexpr int HW = Hh * Ww;          // 36864
static constexpr int Q  = Bn * HW;          // 294912 pixels per channel over batch
static constexpr int NCH = 384;             // 2 paths * 3 branches * 64 channels
#define S_LOG2E 21.64042561333445f          // BETA(=15) * log2(e)
#define EPSBN   1e-5f

// ---------------- workspace layout (floats) ----------------
static constexpr size_t OFF_BSTATS = 0;                       // [2*3*64][2] = 768
static constexpr size_t OFF_YSTATS = 768;                     // [2][2]     = 4
static constexpr size_t OFF_AVEC   = 772;                     // [384]
static constexpr size_t OFF_KC     = 1156;                    // [2]
static constexpr size_t OFF_M      = 1536;                    // [384][Q]  ~432 MB
static constexpr size_t OFF_YRAW   = OFF_M + (size_t)NCH * Q; // [2][Q]

typedef float v2f __attribute__((ext_vector_type(2)));
typedef float v8f __attribute__((ext_vector_type(8)));

// ---------------- kernel 0: zero the small stats region ----------------
__global__ void k_zero(float* __restrict__ ws) {
    int t = blockIdx.x * 256 + threadIdx.x;
    if (t < 772) ws[t] = 0.f;
}

// ---------------- kernel 1: soft morphology (6 branches) + channel stats ---------
// One block = one (b, c, 16x64 tile). LDS tile with halo 3 covers all dilations.
// m is streamed out with TH=NT so the 432MB intermediate doesn't thrash L2.
__global__ __launch_bounds__(256) void k_morph(const float* __restrict__ x,
                                               const float* __restrict__ we,
                                               const float* __restrict__ wd,
                                               float* __restrict__ ws) {
    __shared__ float tile[22 * 70];   // (16+6) x (64+6)
    __shared__ float wsm[54];         // [0..26] erosion w, [27..53] dilation w
    __shared__ float sacc[12];        // 6 branches x {sum, sumsq}

    const int tid = threadIdx.x;
    const int b  = blockIdx.z, c = blockIdx.y;
    const int tx = blockIdx.x % 3, ty = blockIdx.x / 3;   // 3 x 12 tiles
    const int gx0 = tx * 64 - 3, gy0 = ty * 16 - 3;
    const float* xp = x + (size_t)(b * Cc + c) * HW;

    for (int idx = tid; idx < 22 * 70; idx += 256) {
        int ly = idx / 70, lx = idx % 70;
        int gy = gy0 + ly, gx = gx0 + lx;
        float v = 0.f;
        if ((unsigned)gy < (unsigned)Hh && (unsigned)gx < (unsigned)Ww)
            v = xp[gy * Ww + gx];
        tile[idx] = v;
    }
    if (tid < 54) {
        int p  = (tid < 27) ? 0 : 1;
        int t2 = tid - 27 * p;
        int i = t2 / 9, k = t2 % 9;
        const float* wp = p ? wd : we;               // (3, C, 3, 3)
        wsm[tid] = wp[(i * Cc + c) * 9 + k];
    }
    if (tid < 12) sacc[tid] = 0.f;
    __syncthreads();

    const int lx  = tid & 63;     // coalesced along W
    const int row = tid >> 6;     // 0..3
    float a1[6], a2[6];
#pragma unroll
    for (int j = 0; j < 6; ++j) { a1[j] = 0.f; a2[j] = 0.f; }

    float* mbase = ws + OFF_M;
#pragma unroll
    for (int r = 0; r < 4; ++r) {
        int py  = row * 4 + r;                       // 0..15
        int lcy = py + 3, lcx = lx + 3;
        int gy = ty * 16 + py, gx = tx * 64 + lx;
        size_t pix = (size_t)b * HW + (size_t)gy * Ww + gx;
#pragma unroll
        for (int d = 1; d <= 3; ++d) {
            int i = d - 1;
            float t9[9];
#pragma unroll
            for (int ky = 0; ky < 3; ++ky)
#pragma unroll
                for (int kx = 0; kx < 3; ++kx)
                    t9[ky * 3 + kx] = tile[(lcy + (ky - 1) * d) * 70 + (lcx + (kx - 1) * d)];
            float mE = -3.4e38f, mD = -3.4e38f;
#pragma unroll
            for (int k = 0; k < 9; ++k) {
                float zE = wsm[i * 9 + k] - t9[k];
                float zD = wsm[27 + i * 9 + k] + t9[k];
                mE = fmaxf(mE, zE);
                mD = fmaxf(mD, zD);
            }
            float sE = 0.f, sD = 0.f;
#pragma unroll
            for (int k = 0; k < 9; ++k) {
                float zE = wsm[i * 9 + k] - t9[k];
                float zD = wsm[27 + i * 9 + k] + t9[k];
                sE += __builtin_amdgcn_exp2f((zE - mE) * S_LOG2E);   // v_exp_f32
                sD += __builtin_amdgcn_exp2f((zD - mD) * S_LOG2E);
            }
            float oE = -(mE + __builtin_amdgcn_logf(sE) * (1.0f / S_LOG2E)); // v_log_f32
            float oD =  (mD + __builtin_amdgcn_logf(sD) * (1.0f / S_LOG2E));
            // streaming (non-temporal) stores of the 432MB intermediate
            __builtin_nontemporal_store(oE, &mbase[(size_t)(i * Cc + c) * Q + pix]);
            __builtin_nontemporal_store(oD, &mbase[(size_t)(192 + i * Cc + c) * Q + pix]);
            a1[i]     += oE;  a2[i]     += oE * oE;
            a1[3 + i] += oD;  a2[3 + i] += oD * oD;
        }
    }
#pragma unroll
    for (int j = 0; j < 6; ++j) {
        atomicAdd(&sacc[j * 2],     a1[j]);
        atomicAdd(&sacc[j * 2 + 1], a2[j]);
    }
    __syncthreads();
    if (tid < 12) {
        int j = tid >> 1, s = tid & 1;               // j: 0..2 erosion, 3..5 dilation
        int path = j / 3, i = j % 3;
        atomicAdd(&ws[OFF_BSTATS + ((size_t)(path * 3 + i) * Cc + c) * 2 + s], sacc[tid]);
    }
}

// ---------------- kernel 2: fold BN stats into conv coefficients ----------------
__global__ __launch_bounds__(384) void k_prep(const float* __restrict__ bn_ge,
                                              const float* __restrict__ bn_be,
                                              const float* __restrict__ bn_gd,
                                              const float* __restrict__ bn_bd,
                                              const float* __restrict__ cwe,
                                              const float* __restrict__ cwd,
                                              float* __restrict__ ws) {
    __shared__ float kc[2];
    int t = threadIdx.x;                              // 0..383 == global channel
    if (t < 2) kc[t] = 0.f;
    __syncthreads();
    int path = t / 192, r = t % 192;                  // r = i*64 + c
    float inv_n = 1.0f / (float)Q;
    float sum  = ws[OFF_BSTATS + (size_t)t * 2];
    float ssq  = ws[OFF_BSTATS + (size_t)t * 2 + 1];
    float mean = sum * inv_n;
    float var  = ssq * inv_n - mean * mean;
    float rs   = rsqrtf(var + EPSBN);
    float g  = path ? bn_gd[r] : bn_ge[r];
    float be = path ? bn_bd[r] : bn_be[r];
    float w  = path ? cwd[r]  : cwe[r];
    ws[OFF_AVEC + t] = w * g * rs;
    atomicAdd(&kc[path], w * (be - g * mean * rs));
    __syncthreads();
    if (t < 2) ws[OFF_KC + t] = kc[t];
}

// ---------------- kernel 3: 1x1 conv over 384 channels via V_WMMA_F32_16X16X4_F32
// Each wave computes 16 pixels; B-matrix col 0 = erosion coeffs, col 1 = dilation
// coeffs (zero elsewhere), accumulated over 96 K-chunks of 4 channels.
// A-fragment loads are non-temporal (single-use 432MB stream).
__global__ __launch_bounds__(256) void k_conv_wmma(float* __restrict__ ws) {
    __shared__ float sA[NCH];
    __shared__ float sK[2];
    __shared__ float sacc[4];
    int tid = threadIdx.x;
    if (tid < NCH) sA[tid] = ws[OFF_AVEC + tid];
    if (tid < 2)   sK[tid] = ws[OFF_KC + tid];
    if (tid < 4)   sacc[tid] = 0.f;
    __syncthreads();

    int wid  = blockIdx.x * 8 + (tid >> 5);           // 0..18431 (exact cover of Q/16)
    size_t q0 = (size_t)wid * 16;
    int lane  = tid & 31;
    int M     = lane & 15;                            // A row / D column index
    int khalf = lane >> 4;                            // ISA f32 A layout: lanes 16-31 hold K=2,3
    const float* mb = ws + OFF_M;

    v8f acc = {};
    for (int ch4 = 0; ch4 < NCH / 4; ++ch4) {
        int c0 = ch4 * 4 + 2 * khalf;                 // channel for A VGPR0 (VGPR1 = c0+1)
        v2f afrag, bfrag;
        afrag.x = __builtin_nontemporal_load(&mb[(size_t)c0 * Q + q0 + M]);
        afrag.y = __builtin_nontemporal_load(&mb[(size_t)(c0 + 1) * Q + q0 + M]);
        // B[k][n]: VGPR0->k=2*khalf (chan c0), VGPR1->k=2*khalf+1 (chan c0+1); n = M
        float b0 = 0.f, b1 = 0.f;
        int p0 = (c0 < 192) ? 0 : 1;                  // chunk never straddles paths (192%4==0)
        if (M == p0) { b0 = sA[c0]; b1 = sA[c0 + 1]; }
        bfrag.x = b0; bfrag.y = b1;
        acc = __builtin_amdgcn_wmma_f32_16x16x4_f32(false, afrag, false, bfrag,
                                                    (short)0, acc, false, false);
    }
    // D layout: lane<16 -> N=lane, M=v ; lane>=16 -> N=lane-16, M=v+8
    float s1 = 0.f, s2 = 0.f;
    if (M < 2) {
        float kc = sK[M];
        float* yr = ws + OFF_YRAW + (size_t)M * Q;
#pragma unroll
        for (int v = 0; v < 8; ++v) {
            float val = acc[v] + kc;
            yr[q0 + v + 8 * khalf] = val;
            s1 += val; s2 += val * val;
        }
        atomicAdd(&sacc[M * 2],     s1);
        atomicAdd(&sacc[M * 2 + 1], s2);
    }
    __syncthreads();
    if (tid < 4) atomicAdd(&ws[OFF_YSTATS + tid], sacc[tid]);
}

// ---------------- kernel 4: final BN + ReLU + interleave to (B,2,H,W) ----------
__global__ __launch_bounds__(256) void k_final(const float* __restrict__ g_e,
                                               const float* __restrict__ b_e,
                                               const float* __restrict__ g_d,
                                               const float* __restrict__ b_d,
                                               const float* __restrict__ ws,
                                               float* __restrict__ out) {
    int t = blockIdx.x * 256 + threadIdx.x;
    if (t >= 2 * Q) return;
    int path = t / Q;
    int q = t - path * Q;
    int b = q / HW, p = q - b * HW;
    float inv_n = 1.0f / (float)Q;
    float mean = ws[OFF_YSTATS + path * 2] * inv_n;
    float var  = ws[OFF_YSTATS + path * 2 + 1] * inv_n - mean * mean;
    float rs   = rsqrtf(var + EPSBN);
    float g  = path ? g_d[0] : g_e[0];
    float bb = path ? b_d[0] : b_e[0];
    float val = ws[OFF_YRAW + (size_t)path * Q + q];
    float y = g * (val - mean) * rs + bb;
    out[((size_t)b * 2 + path) * HW + p] = fmaxf(y, 0.f);
}

// ---------------- host launch ----------------
extern "C" void kernel_launch(void* const* d_in, const int* in_sizes, int n_in,
                              void* d_out, int out_size, void* d_ws, size_t ws_size,
                              hipStream_t stream) {
    (void)in_sizes; (void)n_in; (void)out_size; (void)ws_size;
    const float* x     = (const float*)d_in[0];
    const float* we    = (const float*)d_in[1];
    const float* wd    = (const float*)d_in[2];
    const float* bn_ge = (const float*)d_in[3];
    const float* bn_be = (const float*)d_in[4];
    const float* bn_gd = (const float*)d_in[5];
    const float* bn_bd = (const float*)d_in[6];
    const float* cwe   = (const float*)d_in[7];
    const float* cwd   = (const float*)d_in[8];
    const float* g_e   = (const float*)d_in[9];
    const float* b_e   = (const float*)d_in[10];
    const float* g_d   = (const float*)d_in[11];
    const float* b_d   = (const float*)d_in[12];
    float* ws  = (float*)d_ws;
    float* out = (float*)d_out;

    k_zero<<<4, 256, 0, stream>>>(ws);
    dim3 g1(36, Cc, Bn);                          // 3x12 tiles, 64 channels, 8 batch
    k_morph<<<g1, 256, 0, stream>>>(x, we, wd, ws);
    k_prep<<<1, 384, 0, stream>>>(bn_ge, bn_be, bn_gd, bn_bd, cwe, cwd, ws);
    k_conv_wmma<<<Q / 16 / 8, 256, 0, stream>>>(ws);   // 2304 blocks, 8 waves each
    k_final<<<(2 * Q + 255) / 256, 256, 0, stream>>>(g_e, b_e, g_d, b_d, ws, out);
}